// PyQComposedBlock_24043226923474
// MI455X (gfx1250) — compile-verified
//
#include <hip/hip_runtime.h>

// ---------------------------------------------------------------------------
// PyQ composed single-qubit gate on a 22-qubit batched state (MI455X / gfx1250)
//
// Stage 1 (compose_u_wmma): U[b] = G7[b]@G6[b]@...@G0[b], b=0..15, via
//   V_WMMA_F32_16X16X4_F32 on block-diagonal 16x16 embeddings (2 halves of 8
//   batches). Accumulator kept transposed (T = U^T) so it is always the A
//   (left) operand: T <- T @ M_k^T. T round-trips through LDS between steps
//   (D-layout -> row-major -> A-layout re-read); M_k^T built directly in B
//   layout from the gate coefficients. One wave, EXEC all ones.
//
// Stage 2 (apply_gate): memory-bound streaming (512 MB @ 23.3 TB/s ~= 22 us).
//   state[a][j][c][b], a=2^10, c=2^11, b=16 innermost. Thread = one float4 of
//   b for one (a,c); loads s_j as b128, 16 FMAs, stores b128. Non-temporal
//   (stream > L2 capacity, zero reuse). Fully coalesced: a wave covers 512 B
//   contiguous per stream.
// ---------------------------------------------------------------------------

typedef __attribute__((ext_vector_type(2))) float v2f;
typedef __attribute__((ext_vector_type(4))) float v4f;
typedef __attribute__((ext_vector_type(8))) float v8f;

#define NQ      22
#define TARGETQ 10
#define BATCH   16
#define NOPS    8
#define ADIM    (1 << TARGETQ)              // 1024
#define CDIM    (1 << (NQ - TARGETQ - 1))   // 2048
#define JSTRIDE (CDIM * BATCH)              // 32768 floats between j=0 and j=1
#define ABLOCK  (2 * CDIM * BATCH)          // 65536 floats per a
#define NELEM   ((1u << NQ) * BATCH)        // 67108864 floats

// ---------------------------------------------------------------------------
// Stage 1: gate-chain composition with f32 WMMA. One block of 32 threads.
// gates: [k][r][cc][b] flat ((k*2+r)*2+cc)*16+b ;  uout: [i][j][b] (64 floats)
// ---------------------------------------------------------------------------
__global__ __launch_bounds__(32) void compose_u_wmma(
    const float* __restrict__ gates, float* __restrict__ uout) {
  __shared__ float T[16 * 16];   // transposed accumulator, row-major
  const int lane = threadIdx.x;  // 0..31, wave32
  const int n    = lane & 15;
  const int half = lane >> 4;

  for (int h = 0; h < 2; ++h) {  // batch halves: b = 8h + t, t=0..7
    // --- init T = M_0^T : T[2t+cc, 2t+r] = gates[0, r, cc, 8h+t] -----------
    for (int e = lane; e < 256; e += 32) {
      const int row = e >> 4, col = e & 15;
      const int t = row >> 1;
      float v = 0.0f;
      if ((col >> 1) == t)
        v = gates[(((col & 1) * 2) + (row & 1)) * 16 + 8 * h + t];
      T[e] = v;
    }
    __syncthreads();

    // --- chain: T <- T @ M_k^T,  k = 1..7 ----------------------------------
    for (int k = 1; k < NOPS; ++k) {
      // A-operand slices (16x4 each, K-chunk q): lane(m=n, half) holds
      //   V0 = T[m, 4q+2*half], V1 = T[m, 4q+2*half+1]  (ISA 7.12.2)
      v2f A[4];
#pragma unroll
      for (int q = 0; q < 4; ++q) {
        const int kk = 4 * q + 2 * half;
        A[q][0] = T[n * 16 + kk];
        A[q][1] = T[n * 16 + kk + 1];
      }
      // B-operand slices (4x16): V0 = row 4q+2*half (cc=0), V1 = row +1 (cc=1)
      // M_k^T[2t+cc, 2t+r] = gates[k, r, cc, 8h+t]; row kk -> t_k=kk>>1,
      // nonzero only at columns n with n>>1 == t_k (r = n&1).
      v8f acc = {};  // C = 0, accumulate K=16 over 4 chained WMMAs
#pragma unroll
      for (int q = 0; q < 4; ++q) {
        const int t_k = 2 * q + half;
        float b0 = 0.0f, b1 = 0.0f;
        if ((n >> 1) == t_k) {
          const int r = n & 1;
          b0 = gates[((k * 2 + r) * 2 + 0) * 16 + 8 * h + t_k];
          b1 = gates[((k * 2 + r) * 2 + 1) * 16 + 8 * h + t_k];
        }
        v2f B;
        B[0] = b0;
        B[1] = b1;
        // (neg_a, A, neg_b, B, c_mod, C, reuse_a, reuse_b)
        acc = __builtin_amdgcn_wmma_f32_16x16x4_f32(
            false, A[q], false, B, (short)0, acc, false, false);
      }
      __syncthreads();
      // D layout: VGPR v, lanes0-15 -> M=v, lanes16-31 -> M=v+8; N = n
#pragma unroll
      for (int v = 0; v < 8; ++v) T[(v + 8 * half) * 16 + n] = acc[v];
      __syncthreads();
    }

    // --- extract: U[i,j,b=8h+t] = U16[2t+i,2t+j] = T[2t+j, 2t+i] -----------
    {
      const int i = lane >> 4, j = (lane >> 3) & 1, t = lane & 7;
      uout[(i * 2 + j) * 16 + 8 * h + t] = T[(2 * t + j) * 16 + (2 * t + i)];
    }
    __syncthreads();
  }
}

// ---------------------------------------------------------------------------
// Stage 2: bandwidth-bound apply. 2^23 threads; thread = one float4 of b.
// ---------------------------------------------------------------------------
__global__ __launch_bounds__(256) void apply_gate(
    const float* __restrict__ state, const float* __restrict__ U,
    float* __restrict__ out) {
  const unsigned tid  = blockIdx.x * blockDim.x + threadIdx.x;
  const unsigned g    = tid & 3;        // which float4 of the 16-wide b axis
  const unsigned site = tid >> 2;       // (a,c) pair index, 0..2^21-1
  const unsigned c    = site & (CDIM - 1);
  const unsigned a    = site >> 11;

  const size_t base = (size_t)a * ABLOCK + (size_t)c * BATCH + (size_t)g * 4;

  const v4f s0 = __builtin_nontemporal_load((const v4f*)(state + base));
  const v4f s1 = __builtin_nontemporal_load((const v4f*)(state + base + JSTRIDE));

  // U[i][j][b]: 64 floats, resident in cache after first block touches it
  const v4f u00 = *(const v4f*)(U + 0 * 16 + g * 4);
  const v4f u01 = *(const v4f*)(U + 1 * 16 + g * 4);
  const v4f u10 = *(const v4f*)(U + 2 * 16 + g * 4);
  const v4f u11 = *(const v4f*)(U + 3 * 16 + g * 4);

  const v4f o0 = u00 * s0 + u01 * s1;   // per-b FMA, i = 0
  const v4f o1 = u10 * s0 + u11 * s1;   // i = 1

  __builtin_nontemporal_store(o0, (v4f*)(out + base));
  __builtin_nontemporal_store(o1, (v4f*)(out + base + JSTRIDE));
}

// ---------------------------------------------------------------------------
extern "C" void kernel_launch(void* const* d_in, const int* in_sizes, int n_in,
                              void* d_out, int out_size, void* d_ws,
                              size_t ws_size, hipStream_t stream) {
  const float* state = (const float*)d_in[0];  // 2^22 * 16 floats
  const float* gates = (const float*)d_in[1];  // 8*2*2*16 = 512 floats
  float* out = (float*)d_out;
  float* U   = (float*)d_ws;                   // 64 floats of scratch

  compose_u_wmma<<<1, 32, 0, stream>>>(gates, U);

  const unsigned threads = NELEM / 8;          // 2 outputs * 4 lanes per thread
  apply_gate<<<threads / 256, 256, 0, stream>>>(state, U, out);
}